// Petri_GCN_21947282882770
// MI455X (gfx1250) — compile-verified
//
#include <hip/hip_runtime.h>

typedef float v2f __attribute__((ext_vector_type(2)));
typedef float v8f __attribute__((ext_vector_type(8)));

#define DH 64
#define NG 512

// ---------------- degree / normalization precompute ----------------

__global__ void k_fill1(float* __restrict__ p, int n) {
  int i = blockIdx.x * blockDim.x + threadIdx.x;
  if (i < n) p[i] = 1.0f;
}

__global__ void k_deg_acc(const long long* __restrict__ ei, const float* __restrict__ w,
                          float* __restrict__ deg, int E) {
  int e = blockIdx.x * blockDim.x + threadIdx.x;
  if (e < E) atomicAdd(&deg[(int)ei[(size_t)E + e]], w[e]);
}

__global__ void k_dinv(float* __restrict__ deg_dinv, float* __restrict__ self_norm, int n) {
  int i = blockIdx.x * blockDim.x + threadIdx.x;
  if (i < n) {
    float di = 1.0f / sqrtf(deg_dinv[i]);
    deg_dinv[i] = di;          // buffer now holds dinv
    self_norm[i] = di * di;
  }
}

__global__ void k_norm(const long long* __restrict__ ei, const float* __restrict__ w,
                       const float* __restrict__ dinv, float* __restrict__ norm, int E) {
  int e = blockIdx.x * blockDim.x + threadIdx.x;
  if (e < E) {
    int s = (int)ei[e];
    int d = (int)ei[(size_t)E + e];
    norm[e] = dinv[s] * w[e] * dinv[d];
  }
}

// ---------------- f32 WMMA GEMM: Out[N x 64] = A[N x 64] @ W[64 x 64] ----------------
// One wave computes a 16-row x 64-col slab: 4 column tiles sharing one A fragment,
// K=64 swept in steps of 4 -> 16*4 = 64 v_wmma_f32_16x16x4_f32 per wave.
// Layout (ISA 7.12.2, f32):
//   A 16x4 : lane half=lane/16, m=lane%16 ; elem v -> A[m][kbase + 2*half + v]
//   B 4x16 : elem v -> B[kbase + 2*half + v][n=lane%16]
//   C 16x16: elem v -> C[v + 8*half][n=lane%16]
__global__ void k_gemm64_wmma(const float* __restrict__ A, const float* __restrict__ W,
                              float* __restrict__ Out, int N) {
  const int wave = threadIdx.x >> 5;
  const int lane = threadIdx.x & 31;
  const int tile = blockIdx.x * (blockDim.x >> 5) + wave;
  const int row0 = tile * 16;
  if (row0 >= N) return;                 // wave-uniform exit: EXEC stays all-1s for live waves
  const int half = lane >> 4;
  const int m = lane & 15;               // row-in-tile for A, col-in-tile for B/C
  int ar = row0 + m;
  if (ar >= N) ar = N - 1;               // clamp (N is a multiple of 16 here; safety only)
  const float* __restrict__ arow = A + (size_t)ar * DH;

  v8f c0 = {}, c1 = {}, c2 = {}, c3 = {};
  #pragma unroll
  for (int k = 0; k < DH; k += 4) {
    const int kk = k + 2 * half;
    v2f a;
    a.x = arow[kk];
    a.y = arow[kk + 1];
    const float* __restrict__ wp = W + (size_t)kk * DH + m;
    v2f b0; b0.x = wp[0];  b0.y = wp[DH + 0];
    v2f b1; b1.x = wp[16]; b1.y = wp[DH + 16];
    v2f b2; b2.x = wp[32]; b2.y = wp[DH + 32];
    v2f b3; b3.x = wp[48]; b3.y = wp[DH + 48];
    c0 = __builtin_amdgcn_wmma_f32_16x16x4_f32(false, a, false, b0, (short)0, c0, false, false);
    c1 = __builtin_amdgcn_wmma_f32_16x16x4_f32(false, a, false, b1, (short)0, c1, false, false);
    c2 = __builtin_amdgcn_wmma_f32_16x16x4_f32(false, a, false, b2, (short)0, c2, false, false);
    c3 = __builtin_amdgcn_wmma_f32_16x16x4_f32(false, a, false, b3, (short)0, c3, false, false);
  }
  #pragma unroll
  for (int v = 0; v < 8; ++v) {
    int r = row0 + v + 8 * half;
    if (r < N) {
      float* __restrict__ orow = Out + (size_t)r * DH + m;
      orow[0]  = c0[v];
      orow[16] = c1[v];
      orow[32] = c2[v];
      orow[48] = c3[v];
    }
  }
}

// ---------------- edge scatter: agg[dst] += norm[e] * hw[src] ----------------
// thread = (edge, channel): channels of one edge are consecutive lanes -> coalesced
// gather from hw[src] and coalesced global_atomic_add_f32 into agg[dst]; both
// arrays (25.6 MB) are L2-resident on MI455X (192 MB L2).
__global__ void k_scatter(const long long* __restrict__ ei, const float* __restrict__ norm,
                          const float* __restrict__ hw, float* __restrict__ agg, int E) {
  int i = blockIdx.x * blockDim.x + threadIdx.x;
  int total = E * DH;
  if (i >= total) return;
  int e = i >> 6;
  int c = i & 63;
  int s = (int)ei[e];
  int d = (int)ei[(size_t)E + e];
  atomicAdd(&agg[(size_t)d * DH + c], norm[e] * hw[(size_t)s * DH + c]);
}

// ---------------- combine: out = [relu](agg + self_norm * hw + b) ----------------
__global__ void k_combine(const float* __restrict__ agg, const float* __restrict__ hw,
                          const float* __restrict__ self_norm, const float* __restrict__ b,
                          float* __restrict__ out, int N, int do_relu) {
  int i = blockIdx.x * blockDim.x + threadIdx.x;
  int total = N * DH;
  if (i >= total) return;
  int n = i >> 6;
  int c = i & 63;
  float v = agg[i] + self_norm[n] * hw[i] + b[c];
  out[i] = do_relu ? fmaxf(v, 0.0f) : v;
}

// ---------------- readout MLP 64->32->1 + scatter into per-graph sums ----------------
__global__ void k_readout(const float* __restrict__ h, const long long* __restrict__ batch,
                          const float* __restrict__ RW1, const float* __restrict__ Rb1,
                          const float* __restrict__ RW2, const float* __restrict__ Rb2,
                          float* __restrict__ sums, float* __restrict__ cnt, int N) {
  __shared__ float sW1[DH * 32];
  __shared__ float sW2[32];
  __shared__ float sb1[32];
  for (int t = threadIdx.x; t < DH * 32; t += blockDim.x) sW1[t] = RW1[t];
  if (threadIdx.x < 32) {
    sW2[threadIdx.x] = RW2[threadIdx.x];
    sb1[threadIdx.x] = Rb1[threadIdx.x];
  }
  __syncthreads();
  int i = blockIdx.x * blockDim.x + threadIdx.x;
  if (i >= N) return;

  const float* __restrict__ hrow = h + (size_t)i * DH;
  float z[32];
  #pragma unroll
  for (int j = 0; j < 32; ++j) z[j] = sb1[j];
  for (int k = 0; k < DH; ++k) {
    float hk = hrow[k];
    const float* __restrict__ wrow = &sW1[k * 32];
    #pragma unroll
    for (int j = 0; j < 32; ++j) z[j] = fmaf(hk, wrow[j], z[j]);  // LDS broadcast reads
  }
  float s = Rb2[0];
  #pragma unroll
  for (int j = 0; j < 32; ++j) s = fmaf(fmaxf(z[j], 0.0f), sW2[j], s);

  int g = (int)batch[i];
  atomicAdd(&sums[g], s);
  atomicAdd(&cnt[g], 1.0f);
}

__global__ void k_final(const float* __restrict__ sums, const float* __restrict__ cnt,
                        float* __restrict__ out, int G) {
  int g = blockIdx.x * blockDim.x + threadIdx.x;
  if (g < G) out[g] = sums[g] / fmaxf(cnt[g], 1.0f);
}

// ---------------- launch ----------------

extern "C" void kernel_launch(void* const* d_in, const int* in_sizes, int n_in,
                              void* d_out, int out_size, void* d_ws, size_t ws_size,
                              hipStream_t stream) {
  (void)n_in; (void)ws_size; (void)out_size;
  const float*     x     = (const float*)d_in[0];
  const long long* ei    = (const long long*)d_in[1];   // [2, E] int64
  const float*     ea    = (const float*)d_in[2];
  const long long* batch = (const long long*)d_in[3];   // [N] int64
  const float* Wl[3] = {(const float*)d_in[4], (const float*)d_in[6], (const float*)d_in[8]};
  const float* bl[3] = {(const float*)d_in[5], (const float*)d_in[7], (const float*)d_in[9]};
  const float* RW1 = (const float*)d_in[10];
  const float* Rb1 = (const float*)d_in[11];
  const float* RW2 = (const float*)d_in[12];
  const float* Rb2 = (const float*)d_in[13];
  float* out = (float*)d_out;

  const int N = in_sizes[0] / DH;
  const int E = in_sizes[2];

  // workspace partition (floats): dinv[N] selfn[N] norm[E] hw[N*64] agg[N*64] hcur[N*64] sums[512] cnt[512]
  float* ws        = (float*)d_ws;
  float* dinv      = ws;  ws += N;            // holds deg first, then dinv
  float* self_norm = ws;  ws += N;
  float* norm      = ws;  ws += E;
  float* hw        = ws;  ws += (size_t)N * DH;
  float* agg       = ws;  ws += (size_t)N * DH;
  float* hcur      = ws;  ws += (size_t)N * DH;
  float* sums      = ws;  ws += NG;
  float* cnt       = ws;

  const int B = 256;
  const int gN  = (N + B - 1) / B;
  const int gE  = (E + B - 1) / B;
  const int gNC = (N * DH + B - 1) / B;
  const int gEC = (E * DH + B - 1) / B;
  const int waveTiles = (N + 15) / 16;
  const int gGemm = (waveTiles + (B / 32) - 1) / (B / 32);

  // normalization precompute
  k_fill1<<<gN, B, 0, stream>>>(dinv, N);                      // deg = 1 (self-loop)
  k_deg_acc<<<gE, B, 0, stream>>>(ei, ea, dinv, E);            // deg += w per dst
  k_dinv<<<gN, B, 0, stream>>>(dinv, self_norm, N);            // dinv, self_norm
  k_norm<<<gE, B, 0, stream>>>(ei, ea, dinv, norm, E);         // per-edge coeff

  // 3 GCN layers
  const float* hin = x;
  for (int layer = 0; layer < 3; ++layer) {
    k_gemm64_wmma<<<gGemm, B, 0, stream>>>(hin, Wl[layer], hw, N);
    hipMemsetAsync(agg, 0, (size_t)N * DH * sizeof(float), stream);
    k_scatter<<<gEC, B, 0, stream>>>(ei, norm, hw, agg, E);
    k_combine<<<gNC, B, 0, stream>>>(agg, hw, self_norm, bl[layer], hcur, N, layer < 2 ? 1 : 0);
    hin = hcur;
  }

  // readout + scatter-mean
  hipMemsetAsync(sums, 0, 2 * NG * sizeof(float), stream);     // sums and cnt are contiguous
  k_readout<<<gN, B, 0, stream>>>(hcur, batch, RW1, Rb1, RW2, Rb2, sums, cnt, N);
  k_final<<<(NG + B - 1) / B, B, 0, stream>>>(sums, cnt, out, NG);
}